// PillarMaxPooling_dense_223338300086
// MI455X (gfx1250) — compile-verified
//
#include <hip/hip_runtime.h>
#include <hip/hip_bf16.h>

typedef __attribute__((ext_vector_type(2))) float v2f;
typedef __attribute__((ext_vector_type(8))) float v8f;

#define CIN   10
#define COUT  64
#define WBEV  512
#define HBEV  512
#define BN_EPS 1e-3f

// ---------------------------------------------------------------- zero fill
__global__ void pp_zero_kernel(float* __restrict__ p, long n) {
    long i = (long)blockIdx.x * blockDim.x + threadIdx.x;
    long stride = (long)gridDim.x * blockDim.x;
    for (; i < n; i += stride) p[i] = 0.0f;
}

// ------------------------------------------------- pass 1: BN batch stats
// thread t: channel c = t&63, quad = t>>6 ; each (block,quad) streams points.
__global__ void pp_stats_kernel(const float* __restrict__ X,
                                const float* __restrict__ W0,
                                float* __restrict__ sums,
                                float* __restrict__ sumsq, int N) {
    int tid = threadIdx.x;
    int c = tid & 63;
    int quad = tid >> 6;
    float w[CIN];
#pragma unroll
    for (int i = 0; i < CIN; ++i) w[i] = W0[i * COUT + c];
    float s0 = 0.f, s1 = 0.f;
    int p = blockIdx.x * 4 + quad;
    int stride = gridDim.x * 4;
    for (; p < N; p += stride) {
        float h = 0.f;
#pragma unroll
        for (int i = 0; i < CIN; ++i) h = fmaf(X[(long)p * CIN + i], w[i], h);
        s0 += h;
        s1 += h * h;
    }
    __shared__ float red[256];
    red[tid] = s0;
    __syncthreads();
    float t0 = 0.f;
    if (quad == 0) t0 = red[c] + red[c + 64] + red[c + 128] + red[c + 192];
    __syncthreads();
    red[tid] = s1;
    __syncthreads();
    if (quad == 0) {
        float t1 = red[c] + red[c + 64] + red[c + 128] + red[c + 192];
        atomicAdd(&sums[c], t0);
        atomicAdd(&sumsq[c], t1);
    }
}

// ----------------------------------------------- pass 2: finalize BN affine
__global__ void pp_finalize_kernel(const float* __restrict__ sums,
                                   const float* __restrict__ sumsq,
                                   const float* __restrict__ gamma,
                                   const float* __restrict__ beta,
                                   float* __restrict__ scale,
                                   float* __restrict__ shift, int N) {
    int c = threadIdx.x;
    if (c < COUT) {
        float inv = 1.0f / (float)N;
        float mu = sums[c] * inv;
        float var = sumsq[c] * inv - mu * mu;
        float rstd = rsqrtf(var + BN_EPS);
        float sc = gamma[c] * rstd;
        scale[c] = sc;
        shift[c] = beta[c] - mu * sc;
    }
}

// ------------------------------------------- pass 3: WMMA s-GEMM + seg-max
// One wave = 16 points. A(16x64=hn), B(64x64=Ws) via v_wmma_f32_16x16x4_f32.
__global__ void pp_max_kernel(const float* __restrict__ X,
                              const int* __restrict__ idx,
                              const float* __restrict__ W0,
                              const float* __restrict__ Ws,
                              const float* __restrict__ bs,
                              const float* __restrict__ scale,
                              const float* __restrict__ shift,
                              float* __restrict__ smax,
                              float* __restrict__ hmax, int N) {
    __shared__ float WsS[COUT * COUT];
    __shared__ float W0S[CIN * COUT];
    __shared__ float scS[COUT], shS[COUT], bsS[COUT];
    int tid = threadIdx.x;
    for (int i = tid; i < COUT * COUT; i += 256) WsS[i] = Ws[i];
    for (int i = tid; i < CIN * COUT; i += 256) W0S[i] = W0[i];
    if (tid < COUT) { scS[tid] = scale[tid]; shS[tid] = shift[tid]; bsS[tid] = bs[tid]; }
    __syncthreads();

    int lane = tid & 31;
    int wave = tid >> 5;
    int hh = lane >> 4;        // half-wave: 0 -> K {0,1}, 1 -> K {2,3}
    int lnh = lane & 15;
    long base = ((long)blockIdx.x * 8 + wave) * 16;
    long p = base + lnh;
    bool pvalid = p < N;
    float x[CIN];
#pragma unroll
    for (int i = 0; i < CIN; ++i) x[i] = pvalid ? X[p * CIN + i] : 0.0f;
    int pid = pvalid ? idx[p] : 0;

    // build A: hn = relu(scale * (x@W0col) + shift), lane layout per ISA 16x4 f32
    v2f a[16];
#pragma unroll
    for (int k = 0; k < 16; ++k) {
        int c0 = 4 * k + 2 * hh;
        float h0 = 0.f, h1 = 0.f;
#pragma unroll
        for (int i = 0; i < CIN; ++i) {
            h0 = fmaf(x[i], W0S[i * COUT + c0], h0);
            h1 = fmaf(x[i], W0S[i * COUT + c0 + 1], h1);
        }
        h0 = fmaxf(fmaf(h0, scS[c0], shS[c0]), 0.f);
        h1 = fmaxf(fmaf(h1, scS[c0 + 1], shS[c0 + 1]), 0.f);
        a[k].x = h0;
        a[k].y = h1;
        if (pvalid) {
            atomicMax((unsigned int*)&hmax[(long)pid * COUT + c0], __float_as_uint(h0));
            atomicMax((unsigned int*)&hmax[(long)pid * COUT + c0 + 1], __float_as_uint(h1));
        }
    }

#pragma unroll
    for (int nt = 0; nt < 4; ++nt) {
        int n0 = nt * 16;
        v8f acc = {};
#pragma unroll
        for (int k = 0; k < 16; ++k) {
            int kr = 4 * k + 2 * hh;
            v2f b;
            b.x = WsS[kr * COUT + n0 + lnh];
            b.y = WsS[(kr + 1) * COUT + n0 + lnh];
            acc = __builtin_amdgcn_wmma_f32_16x16x4_f32(false, a[k], false, b,
                                                        (short)0, acc, false, false);
        }
        int n = n0 + lnh;
        float bsn = bsS[n];
#pragma unroll
        for (int r = 0; r < 8; ++r) {
            int Mrow = r + 8 * hh;
            if (base + Mrow < N) {
                int pidm = __shfl(pid, Mrow, 32);
                float sv = fmaxf(acc[r] + bsn, 0.f);
                atomicMax((unsigned int*)&smax[(long)pidm * COUT + n], __float_as_uint(sv));
            }
        }
    }
}

// -------------------- pass 4: WMMA s-GEMM again + softmax numer/denom sums
__global__ void pp_sum_kernel(const float* __restrict__ X,
                              const int* __restrict__ idx,
                              const float* __restrict__ W0,
                              const float* __restrict__ Ws,
                              const float* __restrict__ bs,
                              const float* __restrict__ scale,
                              const float* __restrict__ shift,
                              const float* __restrict__ smax,
                              float* __restrict__ denom,
                              float* __restrict__ numer, int N) {
    __shared__ float WsS[COUT * COUT];
    __shared__ float W0S[CIN * COUT];
    __shared__ float scS[COUT], shS[COUT], bsS[COUT];
    __shared__ float hnS[8 * 16 * 65];   // per-wave 16x64 hn tile, padded stride 65
    int tid = threadIdx.x;
    for (int i = tid; i < COUT * COUT; i += 256) WsS[i] = Ws[i];
    for (int i = tid; i < CIN * COUT; i += 256) W0S[i] = W0[i];
    if (tid < COUT) { scS[tid] = scale[tid]; shS[tid] = shift[tid]; bsS[tid] = bs[tid]; }
    __syncthreads();

    int lane = tid & 31;
    int wave = tid >> 5;
    int hh = lane >> 4;
    int lnh = lane & 15;
    float* hn = &hnS[wave * 16 * 65];
    long base = ((long)blockIdx.x * 8 + wave) * 16;
    long p = base + lnh;
    bool pvalid = p < N;
    float x[CIN];
#pragma unroll
    for (int i = 0; i < CIN; ++i) x[i] = pvalid ? X[p * CIN + i] : 0.0f;
    int pid = pvalid ? idx[p] : 0;

    v2f a[16];
#pragma unroll
    for (int k = 0; k < 16; ++k) {
        int c0 = 4 * k + 2 * hh;
        float h0 = 0.f, h1 = 0.f;
#pragma unroll
        for (int i = 0; i < CIN; ++i) {
            h0 = fmaf(x[i], W0S[i * COUT + c0], h0);
            h1 = fmaf(x[i], W0S[i * COUT + c0 + 1], h1);
        }
        h0 = fmaxf(fmaf(h0, scS[c0], shS[c0]), 0.f);
        h1 = fmaxf(fmaf(h1, scS[c0 + 1], shS[c0 + 1]), 0.f);
        a[k].x = h0;
        a[k].y = h1;
        hn[lnh * 65 + c0] = h0;          // bank-conflict-free (stride 65)
        hn[lnh * 65 + c0 + 1] = h1;
    }
    __syncthreads();

#pragma unroll
    for (int nt = 0; nt < 4; ++nt) {
        int n0 = nt * 16;
        v8f acc = {};
#pragma unroll
        for (int k = 0; k < 16; ++k) {
            int kr = 4 * k + 2 * hh;
            v2f b;
            b.x = WsS[kr * COUT + n0 + lnh];
            b.y = WsS[(kr + 1) * COUT + n0 + lnh];
            acc = __builtin_amdgcn_wmma_f32_16x16x4_f32(false, a[k], false, b,
                                                        (short)0, acc, false, false);
        }
        int n = n0 + lnh;
        float bsn = bsS[n];
#pragma unroll
        for (int r = 0; r < 8; ++r) {
            int Mrow = r + 8 * hh;
            if (base + Mrow < N) {
                int pidm = __shfl(pid, Mrow, 32);
                long off = (long)pidm * COUT + n;
                float sv = fmaxf(acc[r] + bsn, 0.f);
                float e = __expf(sv - smax[off]);
                atomicAdd(&denom[off], e);
                atomicAdd(&numer[off], hn[Mrow * 65 + n] * e);
            }
        }
    }
}

// ---------------------------------- pass 5: per-pillar finalize + BEV scatter
__global__ void pp_scatter_kernel(const float* __restrict__ numer,
                                  const float* __restrict__ denom,
                                  const float* __restrict__ hmax,
                                  const int* __restrict__ pind,
                                  float* __restrict__ out, int M) {
    long i = (long)blockIdx.x * blockDim.x + threadIdx.x;
    long total = (long)M * COUT;
    if (i >= total) return;
    int p = (int)(i >> 6);
    int c = (int)(i & 63);
    float val = 0.5f * (numer[i] / denom[i] + hmax[i]);
    int b = pind[p * 3 + 0];
    int xq = pind[p * 3 + 1];
    int yq = pind[p * 3 + 2];
    long o = ((long)b * COUT + c) * ((long)WBEV * HBEV) + (long)xq * HBEV + yq;
    out[o] = val;
}

extern "C" void kernel_launch(void* const* d_in, const int* in_sizes, int n_in,
                              void* d_out, int out_size, void* d_ws, size_t ws_size,
                              hipStream_t stream) {
    const float* X     = (const float*)d_in[0];
    const int*   idx   = (const int*)d_in[1];
    const int*   pind  = (const int*)d_in[2];
    const float* W0    = (const float*)d_in[3];
    const float* gamma = (const float*)d_in[4];
    const float* beta  = (const float*)d_in[5];
    const float* Ws    = (const float*)d_in[6];
    const float* bs    = (const float*)d_in[7];
    int N = in_sizes[1];
    int M = in_sizes[2] / 3;
    float* out = (float*)d_out;

    float* ws    = (float*)d_ws;
    float* smax  = ws;
    float* hmaxb = ws + (long)M * COUT;
    float* denom = ws + 2L * M * COUT;
    float* numer = ws + 3L * M * COUT;
    float* sums  = ws + 4L * M * COUT;   // 64
    float* sumsq = sums + 64;            // 64
    float* scale = sums + 128;           // 64
    float* shift = sums + 192;           // 64
    long nws = 4L * M * COUT + 256;

    pp_zero_kernel<<<16384, 256, 0, stream>>>(out, (long)out_size);
    pp_zero_kernel<<<8192, 256, 0, stream>>>(ws, nws);

    pp_stats_kernel<<<256, 256, 0, stream>>>(X, W0, sums, sumsq, N);
    pp_finalize_kernel<<<1, 64, 0, stream>>>(sums, sumsq, gamma, beta, scale, shift, N);

    long tiles = ((long)N + 15) / 16;
    int gemmBlocks = (int)((tiles + 7) / 8);
    pp_max_kernel<<<gemmBlocks, 256, 0, stream>>>(X, idx, W0, Ws, bs, scale, shift,
                                                  smax, hmaxb, N);
    pp_sum_kernel<<<gemmBlocks, 256, 0, stream>>>(X, idx, W0, Ws, bs, scale, shift,
                                                  smax, denom, numer, N);

    long total = (long)M * COUT;
    int scBlocks = (int)((total + 255) / 256);
    pp_scatter_kernel<<<scBlocks, 256, 0, stream>>>(numer, denom, hmaxb, pind, out, M);
}